// HeteroRGCNLayer_15925738733687
// MI455X (gfx1250) — compile-verified
//
#include <hip/hip_runtime.h>
#include <hip/hip_bf16.h>

// ---------------------------------------------------------------------------
// HeteroRGCN layer for MI455X (gfx1250, wave32).
//   h   = feat_src @ W                      (fp32 WMMA 16x16x4)
//   m_e = h[src_e] * (w_e * We + be)        (wave-per-edge, float4/lane)
//   s   = segment_sum(m, dst); deg = segment_sum(1, dst)
//         -> native global_atomic_add_f32 (L2 atomic units, DEV scope)
//   out = relu(0.5 * (inter/deg + intra/deg))
// ---------------------------------------------------------------------------

typedef __attribute__((ext_vector_type(2))) float v2f;
typedef __attribute__((ext_vector_type(8))) float v8f;

#define D_FEAT 128

// Native f32 atomic add, no return (STOREcnt-tracked), device scope.
// Inline asm guarantees the L2 atomic-add path (no CAS loop).
__device__ __forceinline__ void atomic_add_f32_dev(float* p, float v) {
    asm volatile("global_atomic_add_f32 %0, %1, off scope:SCOPE_DEV"
                 :: "v"(p), "v"(v) : "memory");
}

// -------------------------- zero workspace ---------------------------------
__global__ void zero_kernel(float* __restrict__ p, int n) {
    int i = blockIdx.x * blockDim.x + threadIdx.x;
    int stride = gridDim.x * blockDim.x;
    for (; i < n; i += stride) p[i] = 0.0f;
}

// -------------------------- GEMM: H = A @ W --------------------------------
// A: [M,128] row-major, W: [128,128] row-major, H: [M,128].
// Block = 128 threads (4 waves). Block b covers rows [16b,16b+16).
// Wave w computes two 16x16 C tiles at columns 32w and 32w+16, sharing the
// A fragment. K-loop: 32 steps of k=4 -> v_wmma_f32_16x16x4_f32.
__global__ __launch_bounds__(128) void gemm_wmma_kernel(
        const float* __restrict__ A, const float* __restrict__ W,
        float* __restrict__ H, int M) {
    const int lane = threadIdx.x & 31;
    const int wave = threadIdx.x >> 5;
    const int row0 = blockIdx.x * 16;

    // A 16x4 fp32 fragment layout (ISA 7.12.2):
    //   lanes 0-15 : M=lane,    v0=K0, v1=K1
    //   lanes 16-31: M=lane-16, v0=K2, v1=K3
    const int m  = lane & 15;
    const int kh = (lane >> 4) << 1;       // 0 or 2
    const int n  = lane & 15;
    const int n0 = wave * 32;

    int arowi = row0 + m;
    if (arowi >= M) arowi = M - 1;         // safe clamp (M % 16 == 0 here)
    const float* arow = A + (size_t)arowi * D_FEAT;

    v8f c0 = {};
    v8f c1 = {};
    for (int k0 = 0; k0 < D_FEAT; k0 += 4) {
        v2f a, b0, b1;
        a.x = arow[k0 + kh];
        a.y = arow[k0 + kh + 1];
        const float* wr0 = W + (size_t)(k0 + kh) * D_FEAT;
        const float* wr1 = wr0 + D_FEAT;
        // B 4x16 fragment: lanes 0-15 rows K0/K1, lanes 16-31 rows K2/K3
        b0.x = wr0[n0 + n];
        b0.y = wr1[n0 + n];
        b1.x = wr0[n0 + 16 + n];
        b1.y = wr1[n0 + 16 + n];
        c0 = __builtin_amdgcn_wmma_f32_16x16x4_f32(false, a, false, b0,
                                                   (short)0, c0, false, false);
        c1 = __builtin_amdgcn_wmma_f32_16x16x4_f32(false, a, false, b1,
                                                   (short)0, c1, false, false);
    }

    // C layout: VGPR v -> M = v + 8*(lane>=16), N = lane&15
    const int mb = (lane >> 4) << 3;
#pragma unroll
    for (int v = 0; v < 8; ++v) {
        int mm = row0 + mb + v;
        if (mm < M) {
            H[(size_t)mm * D_FEAT + n0 + n]      = c0[v];
            H[(size_t)mm * D_FEAT + n0 + 16 + n] = c1[v];
        }
    }
}

// -------------------------- edge scatter -----------------------------------
// One wave32 per edge; lane handles 4 contiguous channels (float4 = 512B/wave
// contiguous on both the gather h[src] and the atomic scatter to s[dst]).
__global__ __launch_bounds__(256) void edge_kernel(
        const float* __restrict__ h, const int* __restrict__ src,
        const int* __restrict__ dst, const float* __restrict__ w,
        const float* __restrict__ We, const float* __restrict__ be,
        float* __restrict__ s, float* __restrict__ deg, int nedges) {
    const int lane = threadIdx.x & 31;
    const int e = blockIdx.x * (blockDim.x >> 5) + (threadIdx.x >> 5);
    if (e >= nedges) return;

    const int   sv = src[e];
    const int   dv = dst[e];
    const float wv = w[e];

    const int c0 = lane << 2;
    const float4 hv  = *(const float4*)(h  + (size_t)sv * D_FEAT + c0);
    const float4 wef = *(const float4*)(We + c0);
    const float4 bef = *(const float4*)(be + c0);

    float m0 = hv.x * fmaf(wv, wef.x, bef.x);
    float m1 = hv.y * fmaf(wv, wef.y, bef.y);
    float m2 = hv.z * fmaf(wv, wef.z, bef.z);
    float m3 = hv.w * fmaf(wv, wef.w, bef.w);

    float* sp = s + (size_t)dv * D_FEAT + c0;
    atomic_add_f32_dev(sp + 0, m0);
    atomic_add_f32_dev(sp + 1, m1);
    atomic_add_f32_dev(sp + 2, m2);
    atomic_add_f32_dev(sp + 3, m3);
    if (lane == 0) atomic_add_f32_dev(deg + dv, 1.0f);
}

// -------------------------- finalize ---------------------------------------
// out[0] = relu(0.5*(s_rl/deg_rl + s_ll/deg_ll))   (l nodes)
// out[1] = relu(0.5*(s_lr/deg_lr + s_rr/deg_rr))   (r nodes)
__global__ __launch_bounds__(256) void finalize_kernel(
        const float* __restrict__ s_lr, const float* __restrict__ deg_lr,
        const float* __restrict__ s_rl, const float* __restrict__ deg_rl,
        const float* __restrict__ s_ll, const float* __restrict__ deg_ll,
        const float* __restrict__ s_rr, const float* __restrict__ deg_rr,
        float* __restrict__ out, int nd) {
    int i = blockIdx.x * blockDim.x + threadIdx.x;
    if (i >= nd) return;
    const int node = i >> 7;  // D = 128

    const float r_rl = 1.0f / fmaxf(deg_rl[node], 1.0f);
    const float r_ll = 1.0f / fmaxf(deg_ll[node], 1.0f);
    const float r_lr = 1.0f / fmaxf(deg_lr[node], 1.0f);
    const float r_rr = 1.0f / fmaxf(deg_rr[node], 1.0f);

    float ol  = 0.5f * (s_rl[i] * r_rl + s_ll[i] * r_ll);
    float orr = 0.5f * (s_lr[i] * r_lr + s_rr[i] * r_rr);
    out[i]      = fmaxf(ol, 0.0f);
    out[nd + i] = fmaxf(orr, 0.0f);
}

// ---------------------------------------------------------------------------
extern "C" void kernel_launch(void* const* d_in, const int* in_sizes, int n_in,
                              void* d_out, int out_size, void* d_ws, size_t ws_size,
                              hipStream_t stream) {
    const float* feat_l = (const float*)d_in[0];
    const float* feat_r = (const float*)d_in[1];
    const int*   src_lr = (const int*)d_in[2];
    const int*   dst_lr = (const int*)d_in[3];
    const float* w_lr   = (const float*)d_in[4];
    const int*   src_rl = (const int*)d_in[5];
    const int*   dst_rl = (const int*)d_in[6];
    const float* w_rl   = (const float*)d_in[7];
    const int*   src_ll = (const int*)d_in[8];
    const int*   dst_ll = (const int*)d_in[9];
    const float* w_ll   = (const float*)d_in[10];
    const int*   src_rr = (const int*)d_in[11];
    const int*   dst_rr = (const int*)d_in[12];
    const float* w_rr   = (const float*)d_in[13];
    const float* W_lr   = (const float*)d_in[14];
    const float* We_lr  = (const float*)d_in[15];
    const float* be_lr  = (const float*)d_in[16];
    const float* W_ll   = (const float*)d_in[17];
    const float* We_ll  = (const float*)d_in[18];
    const float* be_ll  = (const float*)d_in[19];

    const int n_nodes = in_sizes[0] / D_FEAT;   // 50000
    const int nd      = n_nodes * D_FEAT;       // 6.4M
    const int E_lr = in_sizes[2], E_rl = in_sizes[5];
    const int E_ll = in_sizes[8], E_rr = in_sizes[11];

    // Workspace layout (floats): 4 accumulators, 4 degree vectors, 1 h buffer
    float* ws     = (float*)d_ws;
    float* s_lr   = ws + 0L * nd;
    float* s_rl   = ws + 1L * nd;
    float* s_ll   = ws + 2L * nd;
    float* s_rr   = ws + 3L * nd;
    float* deg_lr = ws + 4L * nd;
    float* deg_rl = deg_lr + n_nodes;
    float* deg_ll = deg_rl + n_nodes;
    float* deg_rr = deg_ll + n_nodes;
    float* h      = deg_rr + n_nodes;

    // 1) zero accumulators + degrees
    const int zero_n = 4 * nd + 4 * n_nodes;
    zero_kernel<<<4096, 256, 0, stream>>>(ws, zero_n);

    const int gemm_grid = (n_nodes + 15) / 16;  // 3125

    // 2) conv lr: inter_r  (feat_l @ W_lr, edge params lr)
    gemm_wmma_kernel<<<gemm_grid, 128, 0, stream>>>(feat_l, W_lr, h, n_nodes);
    edge_kernel<<<(E_lr + 7) / 8, 256, 0, stream>>>(h, src_lr, dst_lr, w_lr,
                                                    We_lr, be_lr, s_lr, deg_lr, E_lr);
    // 3) conv rl: inter_l  (feat_r @ W_lr)
    gemm_wmma_kernel<<<gemm_grid, 128, 0, stream>>>(feat_r, W_lr, h, n_nodes);
    edge_kernel<<<(E_rl + 7) / 8, 256, 0, stream>>>(h, src_rl, dst_rl, w_rl,
                                                    We_lr, be_lr, s_rl, deg_rl, E_rl);
    // 4) conv ll: intra_l  (feat_l @ W_ll)
    gemm_wmma_kernel<<<gemm_grid, 128, 0, stream>>>(feat_l, W_ll, h, n_nodes);
    edge_kernel<<<(E_ll + 7) / 8, 256, 0, stream>>>(h, src_ll, dst_ll, w_ll,
                                                    We_ll, be_ll, s_ll, deg_ll, E_ll);
    // 5) conv rr: intra_r  (feat_r @ W_ll)
    gemm_wmma_kernel<<<gemm_grid, 128, 0, stream>>>(feat_r, W_ll, h, n_nodes);
    edge_kernel<<<(E_rr + 7) / 8, 256, 0, stream>>>(h, src_rr, dst_rr, w_rr,
                                                    We_ll, be_ll, s_rr, deg_rr, E_rr);

    // 6) normalize + combine + relu
    finalize_kernel<<<(nd + 255) / 256, 256, 0, stream>>>(
        s_lr, deg_lr, s_rl, deg_rl, s_ll, deg_ll, s_rr, deg_rr,
        (float*)d_out, nd);
}